// RejectionSampler_65524021068008
// MI455X (gfx1250) — compile-verified
//
#include <hip/hip_runtime.h>
#include <hip/hip_bf16.h>
#include <float.h>

// Problem constants (match the reference).
#define B 64
#define S 8
#define V 32000
#define ROWS (B * S)
#define THREADS 512
#define NWAVES (THREADS / 32)
#define TINY 1.175494350822288e-38f

typedef unsigned int u32x4 __attribute__((ext_vector_type(4)));
typedef int          i32x8 __attribute__((ext_vector_type(8)));
typedef int          i32x4 __attribute__((ext_vector_type(4)));
typedef float        f32x4 __attribute__((ext_vector_type(4)));

// ---------------------------------------------------------------------------
// Tensor Data Mover: DMA one row of V fp32 (128 KB) from global into LDS.
// Descriptor layout per CDNA5 ISA ch.8 (D# groups 0/1; groups 2/3 NULL for a
// 2D tensor). Issued by a single wave; completion via s_wait_tensorcnt.
// ---------------------------------------------------------------------------
__device__ __forceinline__ void tdm_load_row_to_lds(const float* gsrc, void* lds_dst) {
#if __has_builtin(__builtin_amdgcn_tensor_load_to_lds)
  unsigned long long ga = (unsigned long long)gsrc;
  unsigned lds_off = (unsigned)(unsigned long long)lds_dst; // low 32b of LDS aperture = LDS offset

  u32x4 g0;
  g0.x = 1u;                                              // count=1, user mode
  g0.y = lds_off;                                         // lds_addr [63:32]
  g0.z = (unsigned)(ga & 0xFFFFFFFFull);                  // global_addr [95:64]
  g0.w = (unsigned)((ga >> 32) & 0x1FFFFFFull)            // global_addr [120:96]
       | (2u << 30);                                      // type=2 ("image")

  const unsigned dim0 = V, dim1 = 1, tile0 = V, tile1 = 1;
  const unsigned long long str0 = V, str1 = V;
  i32x8 g1;
  g1[0] = 0x00020000;                                     // workgroup_mask=0, data_size=2 (4B)
  g1[1] = (int)((dim0 & 0xFFFFu) << 16);                  // tensor_dim0[15:0] @ bits 63:48
  g1[2] = (int)(((dim0 >> 16) & 0xFFFFu) | ((dim1 & 0xFFFFu) << 16));
  g1[3] = (int)(((dim1 >> 16) & 0xFFFFu) | ((tile0 & 0xFFFFu) << 16));   // tile_dim0
  g1[4] = (int)(tile1 & 0xFFFFu);                         // tile_dim1=1, tile_dim2=0
  g1[5] = (int)(str0 & 0xFFFFFFFFull);                    // tensor_dim0_stride lo
  g1[6] = (int)(((str0 >> 32) & 0xFFFFull) | ((str1 & 0xFFFFull) << 16));
  g1[7] = (int)((str1 >> 16) & 0xFFFFFFFFull);

  i32x4 g2 = {0, 0, 0, 0};
  i32x4 g3 = {0, 0, 0, 0};
#if defined(__clang_major__) && (__clang_major__ >= 23)
  i32x8 g4 = {0, 0, 0, 0, 0, 0, 0, 0};
  __builtin_amdgcn_tensor_load_to_lds(g0, g1, g2, g3, g4, 0);
#else
  __builtin_amdgcn_tensor_load_to_lds(g0, g1, g2, g3, 0);
#endif
#else
  (void)gsrc; (void)lds_dst;
#endif
}

// ---------------------------------------------------------------------------
// Kernel 1: one block per (b,s) row. All streaming is float4 (b128) to reach
// the HBM roofline with minimal issue overhead.
//   Phase A: row max of logits (LDS, ds_load_b128)              -> m
//   Phase B: sum of exp((x-m)/T) (LDS)                          -> Z
//   Phase C: fused: write p = exp((x-m)/T)/Z (non-temporal b128), argmax of
//            max(p - draft_p, tiny)/q  (residual renorm is a positive scale
//            per row -> argmax-invariant, so it is elided), accept test.
// ---------------------------------------------------------------------------
__global__ __launch_bounds__(THREADS) void rs_row_kernel(
    const float* __restrict__ target_logits,   // [ROWS*V]
    const float* __restrict__ temperature,     // [B]
    const int*   __restrict__ draft_token_ids, // [ROWS]
    const float* __restrict__ draft_probs,     // [ROWS*V]
    const float* __restrict__ uniform_probs,   // [ROWS]
    const float* __restrict__ q,               // [B*V]
    float*       __restrict__ probs_out,       // [ROWS*V]
    int*         __restrict__ accept_ws,       // [ROWS]
    int*         __restrict__ rec_ws)          // [ROWS]
{
  extern __shared__ char smem_raw[];
  float* lds_logits = (float*)smem_raw;                          // V floats (128 KB)
  float* redf = (float*)(smem_raw + (size_t)V * sizeof(float));  // NWAVES
  int*   redi = (int*)(redf + NWAVES);                           // NWAVES

  const int row  = blockIdx.x;
  const int b    = row / S;
  const int tid  = threadIdx.x;
  const int lane = tid & 31;
  const int wave = tid >> 5;

  const float* lrow  = target_logits + (size_t)row * V;
  const float* dprow = draft_probs   + (size_t)row * V;
  const float* qrow  = q             + (size_t)b   * V;
  float*       prow  = probs_out     + (size_t)row * V;

#if __has_builtin(__builtin_amdgcn_tensor_load_to_lds)
  if (wave == 0) {
    tdm_load_row_to_lds(lrow, lds_logits);
#if __has_builtin(__builtin_amdgcn_s_wait_tensorcnt)
    __builtin_amdgcn_s_wait_tensorcnt(0);
#endif
  }
#else
  for (int v = tid * 4; v < V; v += THREADS * 4)
    *(f32x4*)&lds_logits[v] = *(const f32x4*)&lrow[v];
#endif

  // Warm L2 with this row's draft_probs while we reduce over LDS.
  for (int v = tid * 32; v < V; v += THREADS * 32)
    __builtin_prefetch(&dprow[v], 0, 1);

  __syncthreads();

  const float invT = 1.0f / temperature[b];

  // ---- Phase A: block max (b128 LDS reads) --------------------------------
  float m = -FLT_MAX;
  for (int v = tid * 4; v < V; v += THREADS * 4) {
    f32x4 x = *(const f32x4*)&lds_logits[v];
    m = fmaxf(m, fmaxf(fmaxf(x.x, x.y), fmaxf(x.z, x.w)));
  }
  for (int off = 16; off > 0; off >>= 1) m = fmaxf(m, __shfl_xor(m, off, 32));
  if (lane == 0) redf[wave] = m;
  __syncthreads();
  {
    float t = redf[lane & (NWAVES - 1)];
    for (int off = NWAVES / 2; off > 0; off >>= 1) t = fmaxf(t, __shfl_xor(t, off, 32));
    m = t;
  }
  __syncthreads();

  // ---- Phase B: block sum of exp ------------------------------------------
  float sum = 0.0f;
  for (int v = tid * 4; v < V; v += THREADS * 4) {
    f32x4 x = *(const f32x4*)&lds_logits[v];
    sum += __expf((x.x - m) * invT);
    sum += __expf((x.y - m) * invT);
    sum += __expf((x.z - m) * invT);
    sum += __expf((x.w - m) * invT);
  }
  for (int off = 16; off > 0; off >>= 1) sum += __shfl_xor(sum, off, 32);
  if (lane == 0) redf[wave] = sum;
  __syncthreads();
  {
    float t = redf[lane & (NWAVES - 1)];
    for (int off = NWAVES / 2; off > 0; off >>= 1) t += __shfl_xor(t, off, 32);
    sum = t;
  }
  __syncthreads();
  const float invSum = 1.0f / sum;

  // ---- Phase C: fused prob write + residual argmax (all b128) -------------
  float bestS = -1.0f;
  int   bestI = 0x7FFFFFFF;
  for (int v = tid * 4; v < V; v += THREADS * 4) {
    f32x4 x = *(const f32x4*)&lds_logits[v];
    f32x4 p;
    p.x = __expf((x.x - m) * invT) * invSum;
    p.y = __expf((x.y - m) * invT) * invSum;
    p.z = __expf((x.z - m) * invT) * invSum;
    p.w = __expf((x.w - m) * invT) * invSum;
    __builtin_nontemporal_store(p, (f32x4*)&prow[v]);            // write-once stream
    f32x4 dp = __builtin_nontemporal_load((const f32x4*)&dprow[v]); // read-once stream
    f32x4 qv = *(const f32x4*)&qrow[v];                          // shared by 8 rows: keep in L2
#pragma unroll
    for (int i = 0; i < 4; ++i) {
      float r  = fmaxf(p[i] - dp[i], TINY);
      float sc = r / qv[i];                                      // Gumbel-max on residual
      if (sc > bestS) { bestS = sc; bestI = v + i; }             // strict > keeps first index
    }
  }
  for (int off = 16; off > 0; off >>= 1) {
    float os = __shfl_xor(bestS, off, 32);
    int   oi = __shfl_xor(bestI, off, 32);
    if (os > bestS || (os == bestS && oi < bestI)) { bestS = os; bestI = oi; }
  }
  if (lane == 0) { redf[wave] = bestS; redi[wave] = bestI; }
  __syncthreads();
  {
    float s2 = redf[lane & (NWAVES - 1)];
    int   i2 = redi[lane & (NWAVES - 1)];
    for (int off = NWAVES / 2; off > 0; off >>= 1) {
      float os = __shfl_xor(s2, off, 32);
      int   oi = __shfl_xor(i2, off, 32);
      if (os > s2 || (os == s2 && oi < i2)) { s2 = os; i2 = oi; }
    }
    bestI = i2;
  }

  if (tid == 0) {
    rec_ws[row] = bestI;
    const int tok = draft_token_ids[row];
    const float ptok  = __expf((lds_logits[tok] - m) * invT) * invSum;
    const float dptok = dprow[tok];
    accept_ws[row] = ((ptok / dptok) >= uniform_probs[row]) ? 1 : 0;
  }
}

// ---------------------------------------------------------------------------
// Kernel 2: per-batch prefix-accept + output assembly ([B, S+1] tokens).
// ---------------------------------------------------------------------------
__global__ __launch_bounds__(64) void rs_finalize_kernel(
    const int* __restrict__ draft_token_ids, // [ROWS]
    const int* __restrict__ bonus_token_ids, // [B]
    const int* __restrict__ accept_ws,       // [ROWS]
    const int* __restrict__ rec_ws,          // [ROWS]
    float*     __restrict__ out_tok)         // [B*(S+1)] (float-encoded tokens)
{
  const int b = threadIdx.x;
  if (b >= B) return;

  int n = 0;
  for (int s = 0; s < S; ++s) {
    if (accept_ws[b * S + s]) n++;
    else break;
  }
  const int fill = (n == S) ? bonus_token_ids[b] : rec_ws[b * S + n];

  for (int pos = 0; pos <= S; ++pos) {
    int val = (pos < n) ? draft_token_ids[b * S + pos]
                        : ((pos == n) ? fill : -1);
    out_tok[b * (S + 1) + pos] = (float)val;
  }
}

// ---------------------------------------------------------------------------
extern "C" void kernel_launch(void* const* d_in, const int* in_sizes, int n_in,
                              void* d_out, int out_size, void* d_ws, size_t ws_size,
                              hipStream_t stream) {
  const float* target_logits   = (const float*)d_in[0];
  const float* temperature     = (const float*)d_in[1];
  // d_in[2] cu_num_draft_tokens: constant (b+1)*S in this setup; not needed.
  const int*   draft_token_ids = (const int*)d_in[3];
  const float* draft_probs     = (const float*)d_in[4];
  const int*   bonus_token_ids = (const int*)d_in[5];
  const float* uniform_probs   = (const float*)d_in[6];
  const float* q               = (const float*)d_in[7];

  float* out_f     = (float*)d_out;            // [0, B*(S+1)) tokens, then probs
  float* probs_out = out_f + B * (S + 1);

  int* accept_ws = (int*)d_ws;                 // [ROWS]
  int* rec_ws    = accept_ws + ROWS;           // [ROWS]

  const size_t smem = (size_t)V * sizeof(float) + NWAVES * (sizeof(float) + sizeof(int)) + 64;

  rs_row_kernel<<<ROWS, THREADS, smem, stream>>>(
      target_logits, temperature, draft_token_ids, draft_probs,
      uniform_probs, q, probs_out, accept_ws, rec_ws);

  rs_finalize_kernel<<<1, 64, 0, stream>>>(
      draft_token_ids, bonus_token_ids, accept_ws, rec_ws, out_f);
}